// ModelNew_4647154615483
// MI455X (gfx1250) — compile-verified
//
#include <hip/hip_runtime.h>
#include <hip/hip_bf16.h>

// MoE SwiGLU (top-2 of 8 experts), fp32, MI455X/gfx1250.
// Sparse routing + grouped GEMMs on the fp32 WMMA path
// (v_wmma_f32_16x16x4_f32). Weight streaming dominates (276 MB fp32, with
// L2 re-reads per token tile), so this version uses a 32-token tile per
// workgroup: halves L2->WGP weight traffic vs M=16 and amortizes every
// B-operand load over two M-tiles (4 WMMAs per B pair in gate/up, 2 in down).

#define HIDDEN  1024
#define INTER   2816
#define NEXP    8
#define TOPK    2
#define NTOK    2048
#define NASSIGN (NTOK * TOPK)   // 4096 (token,slot) assignments
#define MTILE   32              // tokens per workgroup (2 WMMA M-tiles)

typedef __attribute__((ext_vector_type(2))) float v2f;
typedef __attribute__((ext_vector_type(8))) float v8f;

static __device__ inline v8f v8f_zero() {
    v8f z = {0.f, 0.f, 0.f, 0.f, 0.f, 0.f, 0.f, 0.f};
    return z;
}

// ---------------------------------------------------------------------------
// Kernel 1: routing. Wave e scans all assignments in order; ballot + prefix
// popcount builds an order-stable compact list per expert. Deterministic,
// no atomics.
// ---------------------------------------------------------------------------
__global__ void moe_route_k(const int* __restrict__ idx,
                            int* __restrict__ counts,
                            int* __restrict__ lists) {
    const int lane = threadIdx.x & 31;
    const int e    = threadIdx.x >> 5;          // 8 waves = 8 experts
    int cnt = 0;
    int* lst = lists + e * NASSIGN;
    for (int base = 0; base < NASSIGN; base += 32) {
        const int a = base + lane;              // flat (token*TOPK + slot)
        const bool pred = (idx[a] == e);
        const unsigned mask = __builtin_amdgcn_ballot_w32(pred);
        if (pred) {
            const int pos = cnt + __popc(mask & ((1u << lane) - 1u));
            lst[pos] = a;
        }
        cnt += __popc(mask);
    }
    if (lane == 0) counts[e] = cnt;
}

// ---------------------------------------------------------------------------
// Kernel 2: grouped fused SwiGLU MLP per (expert, 32-token tile).
// 8 waves / 256 threads.
//   LDS: sX [32][260]  current 256-wide K panel of the X tile (pad = 4 banks)
//        sH [32][132]  current 128-wide SiLU(g)*u chunk      (pad = 4 banks)
// Per INTER chunk of 128:
//   gate/up: wave w owns 16 INTER cols; two M-tiles share each B pair ->
//            4 WMMAs per (bg,bu) load; K=1024 in steps of 4.
//   down:    wave w owns 128 H cols; A loaded once per k-step feeds all
//            16 WMMAs (8 N-tiles x 2 M-tiles); K=128 chunk.
// Epilogue: scale each token row by its routing weight, scatter to the
// per-assignment slot buffer (each (t,k) written exactly once ->
// bitwise-deterministic combine).
// ---------------------------------------------------------------------------
__global__ __launch_bounds__(256)
void moe_mlp_k(const float* __restrict__ x,
               const float* __restrict__ wflat,     // expert_weights flat [T*K]
               const float* __restrict__ gate,      // [E][I][H]
               const float* __restrict__ up,        // [E][I][H]
               const float* __restrict__ down,      // [E][H][I]
               const int*   __restrict__ counts,
               const int*   __restrict__ lists,
               float*       __restrict__ o_slot) {  // [NASSIGN][H]
    __shared__ float sX[MTILE * 260];
    __shared__ float sH[MTILE * 132];
    __shared__ int   sEnt[MTILE];
    __shared__ float sWt[MTILE];

    const int e    = blockIdx.y;
    const int tile = blockIdx.x;
    const int cnt  = counts[e];
    if (tile * MTILE >= cnt) return;               // uniform early-exit

    const int tid  = threadIdx.x;
    const int lane = tid & 31;
    const int wave = tid >> 5;
    const int half = lane >> 4;                    // K-pair select (WMMA layout)
    const int nl   = lane & 15;                    // M (A) / N (B,C) lane index

    if (tid < MTILE) {
        const int p = tile * MTILE + tid;
        const int a = (p < cnt) ? lists[e * NASSIGN + p] : -1;
        sEnt[tid] = a;
        sWt[tid]  = (a >= 0) ? wflat[a] : 0.0f;
    }
    __syncthreads();

    // X staging mapping: thread = (row 0..31, 32-float segment 0..7)
    const int srow = tid >> 3;
    const int sseg = tid & 7;
    const int sa   = sEnt[srow];
    const float* xrow = x + (size_t)((sa < 0 ? 0 : sa) >> 1) * HIDDEN;

    v8f oacc0[8], oacc1[8];
#pragma unroll
    for (int i = 0; i < 8; ++i) { oacc0[i] = v8f_zero(); oacc1[i] = v8f_zero(); }

    const size_t gu_row = ((size_t)e * INTER + wave * 16 + nl) * HIDDEN + 2 * half;
    const float* dB0 = down + ((size_t)e * HIDDEN + wave * 128 + nl) * INTER + 2 * half;

    for (int ic = 0; ic < INTER / 128; ++ic) {     // 22 chunks
        const float* gB = gate + gu_row + (size_t)(ic * 128) * HIDDEN;
        const float* uB = up   + gu_row + (size_t)(ic * 128) * HIDDEN;
        v8f g0 = v8f_zero(), g1 = v8f_zero();
        v8f u0 = v8f_zero(), u1 = v8f_zero();

        for (int ks = 0; ks < 4; ++ks) {           // K = 1024 in 256-wide panels
            __syncthreads();
            {   // cooperative coalesced stage of X panel -> LDS (b128 stores)
                const float4* src = (const float4*)(xrow + ks * 256 + sseg * 32);
                float4* dst = (float4*)(&sX[srow * 260 + sseg * 32]);
#pragma unroll
                for (int j = 0; j < 8; ++j) dst[j] = src[j];
            }
            __syncthreads();
            const float* gBk = gB + ks * 256;
            const float* uBk = uB + ks * 256;
#pragma unroll 8
            for (int k4 = 0; k4 < 64; ++k4) {      // K step 4
                v2f a0 = *(const v2f*)(&sX[nl * 260 + k4 * 4 + 2 * half]);
                v2f a1 = *(const v2f*)(&sX[(nl + 16) * 260 + k4 * 4 + 2 * half]);
                v2f bg = *(const v2f*)(gBk + k4 * 4);
                v2f bu = *(const v2f*)(uBk + k4 * 4);
                g0 = __builtin_amdgcn_wmma_f32_16x16x4_f32(false, a0, false, bg,
                                                           (short)0, g0, false, false);
                g1 = __builtin_amdgcn_wmma_f32_16x16x4_f32(false, a1, false, bg,
                                                           (short)0, g1, false, false);
                u0 = __builtin_amdgcn_wmma_f32_16x16x4_f32(false, a0, false, bu,
                                                           (short)0, u0, false, false);
                u1 = __builtin_amdgcn_wmma_f32_16x16x4_f32(false, a1, false, bu,
                                                           (short)0, u1, false, false);
            }
        }

        // h = silu(g) * u, elementwise on the C-layout registers
        v8f h0, h1;
#pragma unroll
        for (int r = 0; r < 8; ++r) {
            const float gv0 = g0[r];
            const float gv1 = g1[r];
            h0[r] = (gv0 / (1.0f + __expf(-gv0))) * u0[r];
            h1[r] = (gv1 / (1.0f + __expf(-gv1))) * u1[r];
        }

        __syncthreads();                            // prior sH fully consumed
#pragma unroll
        for (int r = 0; r < 8; ++r) {               // C layout: M = r + 8*half
            sH[(r + 8 * half) * 132 + wave * 16 + nl]        = h0[r];
            sH[(r + 8 * half + 16) * 132 + wave * 16 + nl]   = h1[r];
        }
        __syncthreads();

        // down accumulation for this chunk (K = 128).
        // One A pair per k-step feeds 16 WMMAs; 8 B loads share one base
        // address with immediate offsets (clause-friendly).
        const float* dBc = dB0 + ic * 128;
#pragma unroll 4
        for (int k4 = 0; k4 < 32; ++k4) {
            v2f a0 = *(const v2f*)(&sH[nl * 132 + k4 * 4 + 2 * half]);
            v2f a1 = *(const v2f*)(&sH[(nl + 16) * 132 + k4 * 4 + 2 * half]);
#pragma unroll
            for (int nt = 0; nt < 8; ++nt) {
                v2f bd = *(const v2f*)(dBc + (size_t)(nt * 16) * INTER + k4 * 4);
                oacc0[nt] = __builtin_amdgcn_wmma_f32_16x16x4_f32(false, a0, false, bd,
                                                                  (short)0, oacc0[nt],
                                                                  false, false);
                oacc1[nt] = __builtin_amdgcn_wmma_f32_16x16x4_f32(false, a1, false, bd,
                                                                  (short)0, oacc1[nt],
                                                                  false, false);
            }
        }
    }

    // Epilogue: weight-scale each token row, scatter to its slot buffer row.
#pragma unroll
    for (int nt = 0; nt < 8; ++nt) {
#pragma unroll
        for (int r = 0; r < 8; ++r) {
            const int m0 = r + 8 * half;
            const int m1 = m0 + 16;
            const int a0 = sEnt[m0];
            const int a1 = sEnt[m1];
            const int hcol = wave * 128 + nt * 16 + nl;
            if (a0 >= 0)
                o_slot[(size_t)a0 * HIDDEN + hcol] = oacc0[nt][r] * sWt[m0];
            if (a1 >= 0)
                o_slot[(size_t)a1 * HIDDEN + hcol] = oacc1[nt][r] * sWt[m1];
        }
    }
}

// ---------------------------------------------------------------------------
// Kernel 3: fixed-order combine of the two slot contributions per token.
// float4-vectorized: i4 indexes float4s; 256 float4 per token row.
// ---------------------------------------------------------------------------
__global__ void moe_combine_k(const float4* __restrict__ o_slot4,
                              float4* __restrict__ out4) {
    const int i4 = blockIdx.x * blockDim.x + threadIdx.x;  // over T*H/4
    if (i4 >= NTOK * HIDDEN / 4) return;
    const int t = i4 >> 8;
    const int q = i4 & 255;
    const float4 s0 = o_slot4[(size_t)t * 512 + q];
    const float4 s1 = o_slot4[(size_t)t * 512 + 256 + q];
    float4 r;
    r.x = s0.x + s1.x; r.y = s0.y + s1.y;
    r.z = s0.z + s1.z; r.w = s0.w + s1.w;
    out4[i4] = r;
}

extern "C" void kernel_launch(void* const* d_in, const int* in_sizes, int n_in,
                              void* d_out, int out_size, void* d_ws, size_t ws_size,
                              hipStream_t stream) {
    const float* x    = (const float*)d_in[0];
    const int*   eidx = (const int*)d_in[1];
    const float* ew   = (const float*)d_in[2];
    const float* gate = (const float*)d_in[3];
    const float* up   = (const float*)d_in[4];
    const float* down = (const float*)d_in[5];
    float* out = (float*)d_out;

    // ws layout: counts[8] | lists[8*4096] | o_slot[4096*1024] fp32 (~16.9 MB)
    int*   counts = (int*)d_ws;
    int*   lists  = counts + NEXP;
    float* o_slot = (float*)(lists + NEXP * NASSIGN);

    moe_route_k<<<1, 256, 0, stream>>>(eidx, counts, lists);

    dim3 grid(NASSIGN / MTILE, NEXP);  // worst case 128 tiles/expert; extras exit
    moe_mlp_k<<<grid, 256, 0, stream>>>(x, ew, gate, up, down,
                                        counts, lists, o_slot);

    moe_combine_k<<<(NTOK * HIDDEN / 4 + 255) / 256, 256, 0, stream>>>(
        (const float4*)o_slot, (float4*)out);
}